// GMBLayer_29652454212047
// MI455X (gfx1250) — compile-verified
//
#include <hip/hip_runtime.h>
#include <hip/hip_bf16.h>
#include <math.h>

// ---------------- problem constants ----------------
#define B_G     128
#define N_MAX   64
#define K_HOP   16
#define DIM     128
#define NN      (B_G * N_MAX)       // 8192 nodes
#define EDGES   131072
#define D_STATE 16
#define D_CONV  4
#define DT_RANK 8
#define SEQ     K_HOP               // 16
#define NT      (NN * SEQ)          // 131072 sequence rows

typedef float v2f __attribute__((ext_vector_type(2)));
typedef float v8f __attribute__((ext_vector_type(8)));

#define EP_NONE  0
#define EP_GELU  1
#define EP_RESID 2

// LDS row strides (floats): A panels 16B-aligned + bank-clean for 16-row reads,
// B panels 16B-aligned (reads are lane-consecutive -> conflict free anyway).
#define ASTR 36
#define BSTR 68

__device__ __forceinline__ float gelu_exact(float x) {
    return 0.5f * x * (1.0f + erff(x * 0.70710678118654752f));
}
__device__ __forceinline__ float sigmoidf_(float x) {
    return 1.0f / (1.0f + expf(-x));
}

// Async global->LDS 16B copy (gfx1250 ASYNCcnt pipe). lds_addr: byte offset in
// the wave's LDS aperture (= low 32 bits of the generic shared address).
__device__ __forceinline__ void async_copy_b128(uint32_t lds_addr, const void* gptr) {
    asm volatile("global_load_async_to_lds_b128 %0, %1, off"
                 :: "v"(lds_addr), "v"((uint64_t)(uintptr_t)gptr)
                 : "memory");
}
__device__ __forceinline__ void async_wait_all() {
    asm volatile("s_wait_asynccnt 0x0" ::: "memory");
}
__device__ __forceinline__ uint32_t lds_off(const void* sptr) {
    return (uint32_t)(uintptr_t)sptr;
}

// ---------------------------------------------------------------------------
// Generic fp32 WMMA GEMM:  C[M,Nc] = A[M,K] @ B[K,Nc] (+bias) (+epilogue)
// Call-site invariants: M%64==0, Nc%64==0, K%32==0, lda/ldb%4==0.
// Block = 128 threads (4 waves), 64x64 tile; panels staged by async-to-LDS.
// ---------------------------------------------------------------------------
__global__ __launch_bounds__(128) void gemm_f32_wmma(
    const float* __restrict__ A, int lda,
    const float* __restrict__ B, int ldb,
    float* __restrict__ C, int ldc,
    int K,
    const float* __restrict__ bias,
    const float* __restrict__ resid, int ldr,
    int ep)
{
    __shared__ float As[64 * ASTR];
    __shared__ float Bs[32 * BSTR];

    const int tid   = threadIdx.x;
    const int lane  = tid & 31;
    const int wave  = tid >> 5;
    const int lmod  = lane & 15;
    const int lhalf = lane >> 4;        // 0 or 1
    const long tileM = (long)blockIdx.y * 64;
    const int  tileN = blockIdx.x * 64;

    v8f acc[4] = {};

    for (int k0 = 0; k0 < K; k0 += 32) {
        // stage A panel 64x32 (512 x 16B chunks, 4 per thread) via async pipe
        #pragma unroll
        for (int v = 0; v < 4; ++v) {
            int idx = v * 128 + tid;            // 0..511
            int r   = idx >> 3;                 // row 0..63
            int kk4 = (idx & 7) * 4;            // k offset 0..28
            async_copy_b128(lds_off(&As[r * ASTR + kk4]),
                            A + (tileM + r) * (long)lda + (k0 + kk4));
        }
        // stage B panel 32x64 (512 x 16B chunks) via async pipe
        #pragma unroll
        for (int v = 0; v < 4; ++v) {
            int idx = v * 128 + tid;            // 0..511
            int r   = idx >> 4;                 // k row 0..31
            int c4  = (idx & 15) * 4;           // n offset 0..60
            async_copy_b128(lds_off(&Bs[r * BSTR + c4]),
                            B + (long)(k0 + r) * ldb + tileN + c4);
        }
        async_wait_all();
        __syncthreads();

        const int arow = wave * 16 + lmod;
        #pragma unroll
        for (int kk = 0; kk < 32; kk += 4) {
            v2f a;
            a.x = As[arow * ASTR + kk + 2 * lhalf + 0];
            a.y = As[arow * ASTR + kk + 2 * lhalf + 1];
            #pragma unroll
            for (int nt = 0; nt < 4; ++nt) {
                v2f b;
                b.x = Bs[(kk + 2 * lhalf + 0) * BSTR + nt * 16 + lmod];
                b.y = Bs[(kk + 2 * lhalf + 1) * BSTR + nt * 16 + lmod];
                acc[nt] = __builtin_amdgcn_wmma_f32_16x16x4_f32(
                    false, a, false, b, (short)0, acc[nt], false, false);
            }
        }
        __syncthreads();
    }

    // epilogue: C/D layout -> row = r + 8*lhalf within the wave's 16-row strip
    #pragma unroll
    for (int nt = 0; nt < 4; ++nt) {
        int col = tileN + nt * 16 + lmod;
        float bv = bias ? bias[col] : 0.0f;
        #pragma unroll
        for (int r = 0; r < 8; ++r) {
            long row = tileM + wave * 16 + r + 8 * lhalf;
            float val = acc[nt][r] + bv;
            if (ep == EP_GELU)       val = gelu_exact(val);
            else if (ep == EP_RESID) val += resid[row * (long)ldr + col];
            C[row * (long)ldc + col] = val;
        }
    }
}

// ---------------------------------------------------------------------------
// Batched masked einsum: agg[k,b,i,d] = sum_j mask[b,k,i,j] * dense[b,j,d]
// written directly into flipped layout s_pre[(b*64+i), 15-k, d].
// grid = (2, B_G*K_HOP), block = 128 (4 waves). M=64, K=64, N-tile=64.
// Dense operand staged via async-to-LDS; bool mask converted on the VALU path.
// ---------------------------------------------------------------------------
__global__ __launch_bounds__(128) void agg_wmma(
    const unsigned char* __restrict__ mask,  // [B_G,K_HOP,64,64] bool
    const float* __restrict__ x,             // [B_G,64,128]
    float* __restrict__ spre)                // [NN,16,128]
{
    __shared__ float As[64 * ASTR];
    __shared__ float Bs[64 * BSTR];

    const int tid   = threadIdx.x;
    const int lane  = tid & 31;
    const int wave  = tid >> 5;
    const int lmod  = lane & 15;
    const int lhalf = lane >> 4;
    const int tileN = blockIdx.x * 64;       // 0 or 64
    const int bk    = blockIdx.y;            // 0..2047
    const int b     = bk >> 4;
    const int kh    = bk & 15;

    const unsigned char* mg = mask + ((size_t)bk << 12);      // *4096
    const float* xg = x + (size_t)b * 64 * 128;

    // dense slice 64x64 -> LDS via async pipe (1024 x 16B chunks, 8/thread)
    #pragma unroll
    for (int v = 0; v < 8; ++v) {
        int idx = v * 128 + tid;             // 0..1023
        int r   = idx >> 4;                  // j row 0..63
        int c4  = (idx & 15) * 4;            // d offset 0..60
        async_copy_b128(lds_off(&Bs[r * BSTR + c4]), xg + r * 128 + tileN + c4);
    }
    // mask -> f32 in LDS (full 64x64)
    #pragma unroll
    for (int v = 0; v < 32; ++v) {
        int idx = v * 128 + tid;             // 0..4095
        int r = idx >> 6, c = idx & 63;
        As[r * ASTR + c] = mg[idx] ? 1.0f : 0.0f;
    }
    async_wait_all();
    __syncthreads();

    v8f acc[4] = {};
    const int arow = wave * 16 + lmod;
    #pragma unroll
    for (int kk = 0; kk < 64; kk += 4) {
        v2f a;
        a.x = As[arow * ASTR + kk + 2 * lhalf + 0];
        a.y = As[arow * ASTR + kk + 2 * lhalf + 1];
        #pragma unroll
        for (int nt = 0; nt < 4; ++nt) {
            v2f bb;
            bb.x = Bs[(kk + 2 * lhalf + 0) * BSTR + nt * 16 + lmod];
            bb.y = Bs[(kk + 2 * lhalf + 1) * BSTR + nt * 16 + lmod];
            acc[nt] = __builtin_amdgcn_wmma_f32_16x16x4_f32(
                false, a, false, bb, (short)0, acc[nt], false, false);
        }
    }

    const int t = 15 - kh;
    #pragma unroll
    for (int nt = 0; nt < 4; ++nt) {
        int col = tileN + nt * 16 + lmod;
        #pragma unroll
        for (int r = 0; r < 8; ++r) {
            int i = wave * 16 + r + 8 * lhalf;
            size_t n = (size_t)b * 64 + i;
            spre[n * 2048 + (size_t)t * 128 + col] = acc[nt][r];
        }
    }
}

// ---------------------------------------------------------------------------
// LayerNorm over rows of length 128 (strided in/out). block=128, grid=rows.
// ---------------------------------------------------------------------------
__global__ __launch_bounds__(128) void ln_rows(
    const float* __restrict__ in, long istride,
    float* __restrict__ out, long ostride,
    const float* __restrict__ g, const float* __restrict__ bta)
{
    __shared__ float red[128];
    const long row = blockIdx.x;
    const int d = threadIdx.x;
    float v = in[row * istride + d];
    red[d] = v; __syncthreads();
    for (int s = 64; s > 0; s >>= 1) { if (d < s) red[d] += red[d + s]; __syncthreads(); }
    float m = red[0] * (1.0f / 128.0f);
    __syncthreads();
    float c = v - m;
    red[d] = c * c; __syncthreads();
    for (int s = 64; s > 0; s >>= 1) { if (d < s) red[d] += red[d + s]; __syncthreads(); }
    float var = red[0] * (1.0f / 128.0f);
    out[row * ostride + d] = c * rsqrtf(var + 1e-5f) * g[d] + bta[d];
}

// ---------------------------------------------------------------------------
// Small utility kernels
// ---------------------------------------------------------------------------
__global__ void fill_f32(float* p, float v, int n) {
    int i = blockIdx.x * 256 + threadIdx.x;
    if (i < n) p[i] = v;
}

__global__ void deg_count(const int* __restrict__ ei, float* __restrict__ deg) {
    int e = blockIdx.x * 256 + threadIdx.x;
    if (e < EDGES) atomicAdd(&deg[ei[EDGES + e]], 1.0f);
}

__global__ void rsqrt_deg(const float* __restrict__ deg, float* __restrict__ dis) {
    int i = blockIdx.x * 256 + threadIdx.x;
    if (i < NN) dis[i] = rsqrtf(deg[i] + 1.0f);   // +1 self loop
}

// one 32-lane group per edge, lane handles 4 features
__global__ void edge_scatter(const int* __restrict__ ei,
                             const float* __restrict__ xl,
                             const float* __restrict__ dis,
                             float* __restrict__ acc) {
    size_t gid = (size_t)blockIdx.x * 256 + threadIdx.x;  // EDGES*32 total
    int lane = gid & 31;
    size_t e = gid >> 5;
    if (e >= EDGES) return;
    int r = ei[e], c = ei[EDGES + e];
    float nrm = dis[r] * dis[c];
    const float4 xv = *(const float4*)(xl + (size_t)r * 128 + lane * 4);
    float* dst = acc + (size_t)c * 128 + lane * 4;
    atomicAdd(dst + 0, nrm * xv.x);
    atomicAdd(dst + 1, nrm * xv.y);
    atomicAdd(dst + 2, nrm * xv.z);
    atomicAdd(dst + 3, nrm * xv.w);
}

// local = x + acc + dis^2*xl (self loop) + b_gcn     (in-place into acc)
__global__ void finish_local(const float* __restrict__ x,
                             const float* __restrict__ xl,
                             const float* __restrict__ dis,
                             const float* __restrict__ bg,
                             float* __restrict__ acc) {
    size_t gid = (size_t)blockIdx.x * 256 + threadIdx.x;
    if (gid >= (size_t)NN * 128) return;
    int dd = gid & 127;
    size_t n = gid >> 7;
    float ds = dis[n];
    acc[gid] = x[gid] + acc[gid] + ds * ds * xl[gid] + bg[dd];
}

// depthwise causal conv (D_CONV=4) + SiLU:  in [NN,16,128] -> out [NN,16,128]
__global__ void conv_silu(const float* __restrict__ in,
                          const float* __restrict__ cw,   // [128,1,4]
                          const float* __restrict__ cb,   // [128]
                          float* __restrict__ out) {
    size_t gid = (size_t)blockIdx.x * 256 + threadIdx.x;   // NN*16*128
    if (gid >= (size_t)NN * SEQ * 128) return;
    int d = gid & 127;
    int t = (gid >> 7) & 15;
    size_t n = gid >> 11;
    const float* base = in + n * 2048 + d;
    float a = cb[d];
    #pragma unroll
    for (int w = 0; w < 4; ++w) {
        int ts = t - 3 + w;
        float vv = (ts >= 0) ? base[(size_t)ts * 128] : 0.0f;
        a += cw[d * 4 + w] * vv;
    }
    out[gid] = a * sigmoidf_(a);   // SiLU
}

// ---------------------------------------------------------------------------
// Fused selective scan: per node, runs x_proj + dt_proj + softplus + 16-step
// SSM recurrence, emits gated output at t=15 only.
// block = 128 (thread d owns channel d, state h[16] in registers), grid = NN.
// ---------------------------------------------------------------------------
__global__ __launch_bounds__(128) void scan_kernel(
    const float* __restrict__ xc,        // [NN,16,128] conv output (SiLU'd)
    const float* __restrict__ z15,       // [NN,128]
    const float* __restrict__ x_proj_w,  // [128,40]
    const float* __restrict__ dt_proj_w, // [8,128]
    const float* __restrict__ dt_proj_b, // [128]
    const float* __restrict__ A_log,     // [128,16]
    const float* __restrict__ D_skip,    // [128]
    float* __restrict__ ygated)          // [NN,128]
{
    __shared__ float xrow[128];
    __shared__ float xdbl[40];
    const int n = blockIdx.x;
    const int d = threadIdx.x;

    float Arow[D_STATE];
    #pragma unroll
    for (int s = 0; s < D_STATE; ++s) Arow[s] = -expf(A_log[d * D_STATE + s]);
    float dtw[DT_RANK];
    #pragma unroll
    for (int k = 0; k < DT_RANK; ++k) dtw[k] = dt_proj_w[k * 128 + d];
    const float dtb = dt_proj_b[d];

    float h[D_STATE] = {};
    const float* xcn = xc + (size_t)n * 2048;
    float yout = 0.0f;

    for (int t = 0; t < SEQ; ++t) {
        float u = xcn[(size_t)t * 128 + d];
        xrow[d] = u;
        __syncthreads();
        if (d < 40) {
            float a = 0.0f;
            for (int dd = 0; dd < 128; ++dd) a += xrow[dd] * x_proj_w[dd * 40 + d];
            xdbl[d] = a;
        }
        __syncthreads();

        float dtl = dtb;
        #pragma unroll
        for (int k = 0; k < DT_RANK; ++k) dtl += xdbl[k] * dtw[k];
        float dt = (dtl > 20.0f) ? dtl : log1pf(expf(dtl));   // softplus

        #pragma unroll
        for (int s = 0; s < D_STATE; ++s) {
            float dA = expf(dt * Arow[s]);
            h[s] = dA * h[s] + dt * xdbl[DT_RANK + s] * u;
        }
        if (t == SEQ - 1) {
            float y = 0.0f;
            #pragma unroll
            for (int s = 0; s < D_STATE; ++s) y += h[s] * xdbl[DT_RANK + D_STATE + s];
            y += u * D_skip[d];
            float z = z15[(size_t)n * 128 + d];
            yout = y * (z * sigmoidf_(z));
        }
        __syncthreads();
    }
    ygated[(size_t)n * 128 + d] = yout;
}

// out = local + mA*sigmoid(mB) + xskip0
__global__ void final_combine(const float* __restrict__ local,
                              const float* __restrict__ mA,
                              const float* __restrict__ mB,
                              const float* __restrict__ xskip,
                              float* __restrict__ out) {
    size_t gid = (size_t)blockIdx.x * 256 + threadIdx.x;
    if (gid >= (size_t)NN * 128) return;
    out[gid] = local[gid] + mA[gid] * sigmoidf_(mB[gid]) + xskip[gid];
}

// ---------------------------------------------------------------------------
extern "C" void kernel_launch(void* const* d_in, const int* in_sizes, int n_in,
                              void* d_out, int out_size, void* d_ws, size_t ws_size,
                              hipStream_t stream) {
    (void)in_sizes; (void)n_in; (void)out_size; (void)ws_size;

    const float* x        = (const float*)d_in[0];
    const int*   ei       = (const int*)d_in[1];
    const unsigned char* mask = (const unsigned char*)d_in[3];
    const float* w_gcn    = (const float*)d_in[4];
    const float* b_gcn    = (const float*)d_in[5];
    const float* lnl_g    = (const float*)d_in[6];
    const float* lnl_b    = (const float*)d_in[7];
    const float* m1ln_g   = (const float*)d_in[8];
    const float* m1ln_b   = (const float*)d_in[9];
    const float* m1w1     = (const float*)d_in[10];
    const float* m1b1     = (const float*)d_in[11];
    const float* m1w2     = (const float*)d_in[12];
    const float* m1b2     = (const float*)d_in[13];
    const float* ln_g     = (const float*)d_in[14];
    const float* ln_b     = (const float*)d_in[15];
    const float* in_proj  = (const float*)d_in[16];   // [128,256]
    const float* conv_w   = (const float*)d_in[17];
    const float* conv_b   = (const float*)d_in[18];
    const float* x_proj   = (const float*)d_in[19];
    const float* dtp_w    = (const float*)d_in[20];
    const float* dtp_b    = (const float*)d_in[21];
    const float* A_log    = (const float*)d_in[22];
    const float* Dsk      = (const float*)d_in[23];
    const float* out_proj = (const float*)d_in[24];
    const float* m2w1     = (const float*)d_in[25];
    const float* m2b1     = (const float*)d_in[26];
    const float* m2w2     = (const float*)d_in[27];
    const float* m2b2     = (const float*)d_in[28];
    float* out = (float*)d_out;

    // ---- workspace carving (floats) ----
    const size_t ND = (size_t)NN * 128;           // 1,048,576
    float* w = (float*)d_ws;
    float* lnx   = w;                 w += ND;     // LN buffers / lnagg0 / mB
    float* xl    = w;                 w += ND;     // GCN xl / g1
    float* acc   = w;                 w += ND;     // GCN acc -> local
    float* xskip = w;                 w += ND;     // x_skip3[0]
    float* z15g  = w;                 w += ND;     // z15 -> gmid
    float* yg    = w;                 w += ND;     // ygated -> mA
    float* deg   = w;                 w += 8192;
    float* dis   = w;                 w += 8192;
    float* bufA  = w;                 w += (size_t)NT * 128;  // s_pre -> xc_pre
    float* bufB  = w;                 /* +NT*128 */           // sln -> xc

    const dim3 blk128(128), blk256(256);
    const dim3 gemm8k(2, NN / 64);        // M=8192, Nc=128
    const dim3 gemm131k(2, NT / 64);      // M=131072, Nc=128

    // ---- GCN branch ----
    fill_f32<<<(NN + 255) / 256, blk256, 0, stream>>>(deg, 0.0f, NN);
    deg_count<<<(EDGES + 255) / 256, blk256, 0, stream>>>(ei, deg);
    rsqrt_deg<<<(NN + 255) / 256, blk256, 0, stream>>>(deg, dis);
    ln_rows<<<NN, blk128, 0, stream>>>(x, 128, lnx, 128, lnl_g, lnl_b);
    gemm_f32_wmma<<<gemm8k, blk128, 0, stream>>>(lnx, 128, w_gcn, 128, xl, 128, 128,
                                                 nullptr, nullptr, 0, EP_NONE);
    fill_f32<<<(int)((ND + 255) / 256), blk256, 0, stream>>>(acc, 0.0f, (int)ND);
    edge_scatter<<<(EDGES * 32) / 256, blk256, 0, stream>>>(ei, xl, dis, acc);
    finish_local<<<(int)((ND + 255) / 256), blk256, 0, stream>>>(x, xl, dis, b_gcn, acc);

    // ---- masked hop aggregation -> s_pre (flipped) ----
    agg_wmma<<<dim3(2, B_G * K_HOP), blk128, 0, stream>>>(mask, x, bufA);

    // ---- MLP1 branch (only agg[0] == s_pre[:,15,:] matters) ----
    ln_rows<<<NN, blk128, 0, stream>>>(bufA + 15 * 128, 2048, lnx, 128, m1ln_g, m1ln_b);
    gemm_f32_wmma<<<gemm8k, blk128, 0, stream>>>(lnx, 128, m1w1, 128, xl, 128, 128,
                                                 m1b1, nullptr, 0, EP_GELU);
    gemm_f32_wmma<<<gemm8k, blk128, 0, stream>>>(xl, 128, m1w2, 128, xskip, 128, 128,
                                                 m1b2, bufA + 15 * 128, 2048, EP_RESID);

    // ---- Mamba branch ----
    ln_rows<<<NT, blk128, 0, stream>>>(bufA, 128, bufB, 128, ln_g, ln_b);
    // z gate needed only at t=15
    gemm_f32_wmma<<<gemm8k, blk128, 0, stream>>>(bufB + 15 * 128, 2048,
                                                 in_proj + 128, 256, z15g, 128, 128,
                                                 nullptr, nullptr, 0, EP_NONE);
    // xc pre-conv for all timesteps
    gemm_f32_wmma<<<gemm131k, blk128, 0, stream>>>(bufB, 128, in_proj, 256, bufA, 128, 128,
                                                   nullptr, nullptr, 0, EP_NONE);
    conv_silu<<<(int)(((size_t)NT * 128 + 255) / 256), blk256, 0, stream>>>(bufA, conv_w,
                                                                            conv_b, bufB);
    scan_kernel<<<NN, blk128, 0, stream>>>(bufB, z15g, x_proj, dtp_w, dtp_b, A_log, Dsk, yg);

    // out_proj with fused GELU -> g
    gemm_f32_wmma<<<gemm8k, blk128, 0, stream>>>(yg, 128, out_proj, 128, z15g, 128, 128,
                                                 nullptr, nullptr, 0, EP_GELU);
    // MLP2 gate pair
    gemm_f32_wmma<<<gemm8k, blk128, 0, stream>>>(z15g, 128, m2w1, 128, yg, 128, 128,
                                                 m2b1, nullptr, 0, EP_NONE);
    gemm_f32_wmma<<<gemm8k, blk128, 0, stream>>>(z15g, 128, m2w2, 128, lnx, 128, 128,
                                                 m2b2, nullptr, 0, EP_NONE);

    final_combine<<<(int)((ND + 255) / 256), blk256, 0, stream>>>(acc, yg, lnx, xskip, out);
}